// NeiVar_5643587027585
// MI455X (gfx1250) — compile-verified
//
#include <hip/hip_runtime.h>
#include <hip/hip_bf16.h>

typedef float v2f __attribute__((ext_vector_type(2)));
typedef float v8f __attribute__((ext_vector_type(8)));

#define IN_DIM 128
#define EMB    64

// ---------------------------------------------------------------------------
// Zero a float region (accumulators must be re-zeroed every call: the harness
// poisons workspace once and replays the graph).
// ---------------------------------------------------------------------------
__global__ void nv_zero_kernel(float* __restrict__ p, int count) {
    int i = blockIdx.x * blockDim.x + threadIdx.x;
    if (i < count) p[i] = 0.0f;
}

// ---------------------------------------------------------------------------
// h = x @ W + b via V_WMMA_F32_16X16X4_F32.
// One wave computes one 16x16 tile of h. B (W tile) preloaded into 32 v2f
// registers; K-loop issues 32 WMMA ops (K=4 each -> K=128 total).
// A 16x4 layout: lanes 0-15 hold M=lane, K={0,1}; lanes 16-31 M=lane-16, K={2,3}.
// B 4x16 layout: VGPR0 = rows K=0 (lanes 0-15) / K=2 (lanes 16-31), VGPR1 = K=1/K=3.
// C 16x16: VGPR v = row v (lanes 0-15) / row v+8 (lanes 16-31), N = lane&15.
// ---------------------------------------------------------------------------
__global__ void nv_gemm_wmma_kernel(const float* __restrict__ x,
                                    const float* __restrict__ W,
                                    const float* __restrict__ bias,
                                    float* __restrict__ h,
                                    int n_nodes) {
    const int lane  = threadIdx.x & 31;
    const int wave  = blockIdx.x * (blockDim.x >> 5) + (threadIdx.x >> 5);
    const int tile_m = wave >> 2;          // EMB/16 == 4 column tiles
    const int tile_n = wave & 3;
    const int row0 = tile_m * 16;
    if (row0 >= n_nodes) return;
    const int col0 = tile_n * 16;
    const int half = lane >> 4;            // 0: K lo pair, 1: K hi pair
    const int l15  = lane & 15;

    // Preload B tile: 32 K-steps of v2f per lane.
    v2f Breg[32];
#pragma unroll
    for (int kk = 0; kk < 32; ++kk) {
        const int kbase = kk * 4 + half * 2;
        Breg[kk].x = W[(kbase + 0) * EMB + col0 + l15];
        Breg[kk].y = W[(kbase + 1) * EMB + col0 + l15];
    }

    v8f c = {};
    const float* __restrict__ arow = x + (size_t)(row0 + l15) * IN_DIM;
#pragma unroll
    for (int kk = 0; kk < 32; ++kk) {
        const int kbase = kk * 4 + half * 2;
        v2f a;
        a.x = arow[kbase + 0];             // contiguous pair -> b64 load
        a.y = arow[kbase + 1];
        c = __builtin_amdgcn_wmma_f32_16x16x4_f32(
                /*neg_a=*/false, a, /*neg_b=*/false, Breg[kk],
                /*c_mod=*/(short)0, c, /*reuse_a=*/false, /*reuse_b=*/false);
    }

    const float bv = bias[col0 + l15];
#pragma unroll
    for (int v = 0; v < 8; ++v) {
        const int m = v + half * 8;
        h[(size_t)(row0 + m) * EMB + col0 + l15] = c[v] + bv;
    }
}

// ---------------------------------------------------------------------------
// Row L2-normalize: one wave32 per node, 2 floats per lane, shuffle reduce.
// ---------------------------------------------------------------------------
__global__ void nv_l2norm_kernel(float* __restrict__ h, int n_nodes) {
    const int wave = blockIdx.x * (blockDim.x >> 5) + (threadIdx.x >> 5);
    const int lane = threadIdx.x & 31;
    if (wave >= n_nodes) return;
    float* p = h + (size_t)wave * EMB + lane * 2;
    float2 v = *(const float2*)p;
    float ss = v.x * v.x + v.y * v.y;
#pragma unroll
    for (int off = 16; off > 0; off >>= 1)
        ss += __shfl_xor(ss, off, 32);
    const float inv = 1.0f / sqrtf(ss);
    v.x *= inv; v.y *= inv;
    *(float2*)p = v;
}

// ---------------------------------------------------------------------------
// In-degree (float so the later divides need no conversion).
// ---------------------------------------------------------------------------
__global__ void nv_degree_kernel(const int* __restrict__ dst,
                                 float* __restrict__ deg, int n_edges) {
    int e = blockIdx.x * blockDim.x + threadIdx.x;
    if (e < n_edges) atomicAdd(&deg[dst[e]], 1.0f);
}

// ---------------------------------------------------------------------------
// Scatter-sum of h[src] into msum[dst]. 16 threads/edge, float4 per thread.
// ---------------------------------------------------------------------------
__global__ void nv_mean_scatter_kernel(const float* __restrict__ h,
                                       const int* __restrict__ src,
                                       const int* __restrict__ dst,
                                       float* __restrict__ msum, int n_edges) {
    const int tid = blockIdx.x * blockDim.x + threadIdx.x;
    const int e = tid >> 4;
    if (e >= n_edges) return;
    const int q = tid & 15;
    const int s = src[e];
    const int d = dst[e];
    if (e + 64 < n_edges)  // prefetch a future source row (global_prefetch_b8)
        __builtin_prefetch(h + (size_t)src[e + 64] * EMB, 0, 3);
    const float4 hv = *(const float4*)(h + (size_t)s * EMB + q * 4);
    float* out = msum + (size_t)d * EMB + q * 4;
    atomicAdd(out + 0, hv.x);
    atomicAdd(out + 1, hv.y);
    atomicAdd(out + 2, hv.z);
    atomicAdd(out + 3, hv.w);
}

// ---------------------------------------------------------------------------
// msum -> mean in place: divide by max(deg, 1).
// ---------------------------------------------------------------------------
__global__ void nv_mean_fin_kernel(float* __restrict__ msum,
                                   const float* __restrict__ deg, int count) {
    int i = blockIdx.x * blockDim.x + threadIdx.x;
    if (i >= count) return;
    const float dg = deg[i >> 6];
    msum[i] = msum[i] / fmaxf(dg, 1.0f);
}

// ---------------------------------------------------------------------------
// Variance scatter: sum_d (h[src]-mean[dst])^2 accumulated as ONE scalar
// atomic per (edge, 4-dim chunk) since output sums over features anyway.
// ---------------------------------------------------------------------------
__global__ void nv_var_scatter_kernel(const float* __restrict__ h,
                                      const float* __restrict__ mean,
                                      const int* __restrict__ src,
                                      const int* __restrict__ dst,
                                      float* __restrict__ varsum, int n_edges) {
    const int tid = blockIdx.x * blockDim.x + threadIdx.x;
    const int e = tid >> 4;
    if (e >= n_edges) return;
    const int q = tid & 15;
    const int s = src[e];
    const int d = dst[e];
    if (e + 64 < n_edges)
        __builtin_prefetch(h + (size_t)src[e + 64] * EMB, 0, 3);
    const float4 hv = *(const float4*)(h + (size_t)s * EMB + q * 4);
    const float4 mv = *(const float4*)(mean + (size_t)d * EMB + q * 4);
    const float dx = hv.x - mv.x, dy = hv.y - mv.y,
                dz = hv.z - mv.z, dw = hv.w - mv.w;
    atomicAdd(&varsum[d], dx * dx + dy * dy + dz * dz + dw * dw);
}

// ---------------------------------------------------------------------------
// out[i] = varsum[i] / max(deg[i], 1)
// ---------------------------------------------------------------------------
__global__ void nv_out_kernel(const float* __restrict__ varsum,
                              const float* __restrict__ deg,
                              float* __restrict__ out, int n_nodes) {
    int i = blockIdx.x * blockDim.x + threadIdx.x;
    if (i < n_nodes) out[i] = varsum[i] / fmaxf(deg[i], 1.0f);
}

extern "C" void kernel_launch(void* const* d_in, const int* in_sizes, int n_in,
                              void* d_out, int out_size, void* d_ws, size_t ws_size,
                              hipStream_t stream) {
    const float* x    = (const float*)d_in[0];   // [N, 128]
    const int*   ei   = (const int*)d_in[1];     // [2, E] (JAX x64-off -> int32)
    const float* W    = (const float*)d_in[2];   // [128, 64]
    const float* bias = (const float*)d_in[3];   // [64]
    float* out = (float*)d_out;

    const int n = in_sizes[0] / IN_DIM;          // 50000
    const int E = in_sizes[1] / 2;               // 800000
    const int* src = ei;
    const int* dst = ei + E;

    // Workspace layout (floats): h[N*64] | msum[N*64] | deg[N] | varsum[N]
    float* h      = (float*)d_ws;
    float* msum   = h + (size_t)n * EMB;
    float* deg    = msum + (size_t)n * EMB;
    float* varsum = deg + n;

    // 1) zero accumulators (msum, deg, varsum are contiguous)
    {
        const int count = n * EMB + 2 * n;
        nv_zero_kernel<<<(count + 255) / 256, 256, 0, stream>>>(msum, count);
    }
    // 2) GEMM h = x@W + b  (WMMA, one wave per 16x16 tile)
    {
        const int waves = ((n + 15) / 16) * (EMB / 16);
        const int blocks = (waves * 32 + 255) / 256;
        nv_gemm_wmma_kernel<<<blocks, 256, 0, stream>>>(x, W, bias, h, n);
    }
    // 3) L2 normalize rows (one wave per node)
    nv_l2norm_kernel<<<(n + 7) / 8, 256, 0, stream>>>(h, n);
    // 4) in-degree
    nv_degree_kernel<<<(E + 255) / 256, 256, 0, stream>>>(dst, deg, E);
    // 5) scatter-sum of neighbor features
    {
        const long long total = (long long)E * 16;
        nv_mean_scatter_kernel<<<(int)((total + 255) / 256), 256, 0, stream>>>(
            h, src, dst, msum, E);
    }
    // 6) msum -> mean
    nv_mean_fin_kernel<<<(n * EMB + 255) / 256, 256, 0, stream>>>(msum, deg, n * EMB);
    // 7) variance scatter (scalar atomic per edge-chunk)
    {
        const long long total = (long long)E * 16;
        nv_var_scatter_kernel<<<(int)((total + 255) / 256), 256, 0, stream>>>(
            h, msum, src, dst, varsum, E);
    }
    // 8) finalize
    nv_out_kernel<<<(n + 255) / 256, 256, 0, stream>>>(varsum, deg, out, n);
}